// ClassCapsLayer_19473381720251
// MI455X (gfx1250) — compile-verified
//
#include <hip/hip_runtime.h>
#include <hip/hip_bf16.h>
#include <math.h>

// Sizes fixed by the reference
#define B_ 8
#define C_ 32
#define R_ 2048
#define I_ 64
#define O_ 64

typedef float v2f __attribute__((ext_vector_type(2)));
typedef float v8f __attribute__((ext_vector_type(8)));

// ---------------------------------------------------------------------------
// Kernel 1: priors[b][c][r][o] = sum_i x[b][c][r][i] * W[c][r][i][o]
// One wave32 per (c,r): M=16 (8 valid batch rows), N=64 (4 tiles), K=64
// (16 chunks of 4) using V_WMMA_F32_16X16X4_F32.  Memory-bound: ~1.2 GB read.
// ---------------------------------------------------------------------------
__global__ __launch_bounds__(256) void caps_priors_wmma(
    const float* __restrict__ x,     // [B][C][R][I]
    const float* __restrict__ rw,    // [C][R][I][O]
    float* __restrict__ priors)      // [B][C][R][O]
{
    const int lane  = threadIdx.x & 31;
    const int wave  = threadIdx.x >> 5;
    const int pair  = blockIdx.x * 8 + wave;      // 0 .. C_*R_-1
    const int c     = pair >> 11;                 // / R_
    const int r     = pair & (R_ - 1);

    const int mrow  = lane & 15;                  // A-matrix row M (batch b)
    const int khalf = lane >> 4;                  // 0: K pair {0,1}; 1: {2,3}
    const int ncol  = lane & 15;                  // B/D column within tile

    // Zero out A rows M=8..15 without EXEC divergence (WMMA needs EXEC=all 1s)
    const float mscale = (mrow < 8) ? 1.0f : 0.0f;
    const float* xb = x + (((size_t)(mrow & 7) * C_ + c) * R_ + r) * I_;
    const float* Wt = rw + ((size_t)c * R_ + r) * (I_ * O_);

    v8f acc0 = {}, acc1 = {}, acc2 = {}, acc3 = {};

    #pragma unroll 4
    for (int k = 0; k < 16; ++k) {
        const int i0 = 4 * k + 2 * khalf;
        v2f a;
        a[0] = xb[i0 + 0] * mscale;               // A vgpr0: K = i0
        a[1] = xb[i0 + 1] * mscale;               // A vgpr1: K = i0+1
        // B fragment rows: vgpr v holds K = 4k + v + 2*khalf
        const float* wk0 = Wt + (size_t)(i0 + 0) * O_ + ncol;
        const float* wk1 = Wt + (size_t)(i0 + 1) * O_ + ncol;
        v2f b0, b1, b2, b3;
        b0[0] = wk0[ 0]; b0[1] = wk1[ 0];
        b1[0] = wk0[16]; b1[1] = wk1[16];
        b2[0] = wk0[32]; b2[1] = wk1[32];
        b3[0] = wk0[48]; b3[1] = wk1[48];
        acc0 = __builtin_amdgcn_wmma_f32_16x16x4_f32(false, a, false, b0, (short)0, acc0, false, false);
        acc1 = __builtin_amdgcn_wmma_f32_16x16x4_f32(false, a, false, b1, (short)0, acc1, false, false);
        acc2 = __builtin_amdgcn_wmma_f32_16x16x4_f32(false, a, false, b2, (short)0, acc2, false, false);
        acc3 = __builtin_amdgcn_wmma_f32_16x16x4_f32(false, a, false, b3, (short)0, acc3, false, false);
    }

    // D layout: vgpr j, lanes 0-15 -> M=j (valid b), lanes 16-31 -> M=j+8 (pad)
    if (lane < 16) {
        #pragma unroll
        for (int j = 0; j < 8; ++j) {
            size_t base = (((size_t)j * C_ + c) * R_ + r) * O_ + lane;
            priors[base +  0] = acc0[j];
            priors[base + 16] = acc1[j];
            priors[base + 32] = acc2[j];
            priors[base + 48] = acc3[j];
        }
    }
}

// ---------------------------------------------------------------------------
// Kernel 2: routing reduction over R.
//  mode 0: s[b,c,o] = mean_r priors  (softmax of zero logits is uniform)
//  mode 1: online softmax of logits = priors * w[b,c,o]; s = sum probs*priors
// One wave handles 32 consecutive o for a (b,c): coalesced 128B reads.
// ---------------------------------------------------------------------------
__global__ __launch_bounds__(256) void caps_route_iter(
    const float* __restrict__ priors,   // [B][C][R][O]
    const float* __restrict__ wcoef,    // [B][C][O] cumulative sum of v's
    float* __restrict__ s_out,          // [B][C][O]
    int softmax_mode)
{
    const int gwave = (blockIdx.x * blockDim.x + threadIdx.x) >> 5; // 0..511
    const int lane  = threadIdx.x & 31;
    const int og    = gwave & 1;          // which half of O
    const int bc    = gwave >> 1;         // b*C_ + c
    const int o     = og * 32 + lane;

    const float* p = priors + (size_t)bc * R_ * O_ + o;

    if (!softmax_mode) {
        float acc = 0.0f;
        for (int r = 0; r < R_; ++r) acc += p[(size_t)r * O_];
        s_out[bc * O_ + o] = acc * (1.0f / (float)R_);
    } else {
        const float w = wcoef[bc * O_ + o];
        float m = -3.402823466e38f, d = 0.0f, num = 0.0f;
        for (int r = 0; r < R_; ++r) {
            const float pr = p[(size_t)r * O_];
            const float l  = pr * w;
            const float mn = fmaxf(m, l);
            const float sc = __expf(m - mn);   // rescale old accum
            const float e  = __expf(l - mn);
            d   = d * sc + e;
            num = num * sc + e * pr;
            m   = mn;
        }
        s_out[bc * O_ + o] = num / d;
    }
}

// ---------------------------------------------------------------------------
// Kernel 3: squash with GLOBAL Frobenius norm (faithful to torch.norm(s)):
//   n2 = sum over ALL 16384 elements of s^2
//   v  = (n2/(1+n2)) * s / sqrt(n2)
// Single 1024-thread block; maintains w (running sum of v) and final output.
// ---------------------------------------------------------------------------
__global__ __launch_bounds__(1024) void caps_squash(
    const float* __restrict__ s,
    float* __restrict__ wcoef,
    float* __restrict__ vout,
    int init_w, int write_out)
{
    __shared__ float red[1024];
    const int t = threadIdx.x;
    const int N = B_ * C_ * O_;           // 16384

    float local = 0.0f;
    for (int i = t; i < N; i += 1024) { const float sv = s[i]; local += sv * sv; }
    red[t] = local;
    __syncthreads();
    for (int off = 512; off > 0; off >>= 1) {
        if (t < off) red[t] += red[t + off];
        __syncthreads();
    }
    const float n2    = red[0];
    const float scale = (n2 / (1.0f + n2)) * rsqrtf(n2);

    for (int i = t; i < N; i += 1024) {
        const float v = scale * s[i];
        if (write_out) {
            vout[i] = v;
        } else if (init_w) {
            wcoef[i] = v;                 // w = v0 (overwrites poison)
        } else {
            wcoef[i] += v;                // w = v0 + v1
        }
    }
}

// ---------------------------------------------------------------------------
extern "C" void kernel_launch(void* const* d_in, const int* in_sizes, int n_in,
                              void* d_out, int out_size, void* d_ws, size_t ws_size,
                              hipStream_t stream) {
    (void)in_sizes; (void)n_in; (void)out_size; (void)ws_size;
    const float* x  = (const float*)d_in[0];   // [8,32,2048,1,64]
    const float* rw = (const float*)d_in[1];   // [32,2048,64,64]
    // d_in[2] = iterations (==3); loop is unrolled below.
    float* vout = (float*)d_out;               // [8,32,1,1,64] = 16384 f32

    // Workspace layout: priors (128 MB) | s (64 KB) | w (64 KB)
    float* priors = (float*)d_ws;
    float* s      = priors + (size_t)B_ * C_ * R_ * O_;
    float* wcoef  = s + B_ * C_ * O_;

    // priors = x @ route_weights  (65536 GEMMs of [8x64]x[64x64], WMMA f32)
    caps_priors_wmma<<<(C_ * R_) / 8, 256, 0, stream>>>(x, rw, priors);

    // iteration 0: uniform probs -> mean; v0 = squash(s); w = v0
    caps_route_iter<<<64, 256, 0, stream>>>(priors, wcoef, s, 0);
    caps_squash<<<1, 1024, 0, stream>>>(s, wcoef, vout, /*init_w=*/1, /*write_out=*/0);

    // iteration 1: logits = priors*w; v1 = squash(s); w += v1
    caps_route_iter<<<64, 256, 0, stream>>>(priors, wcoef, s, 1);
    caps_squash<<<1, 1024, 0, stream>>>(s, wcoef, vout, 0, 0);

    // iteration 2 (final): logits = priors*w; out = squash(s)
    caps_route_iter<<<64, 256, 0, stream>>>(priors, wcoef, s, 1);
    caps_squash<<<1, 1024, 0, stream>>>(s, wcoef, vout, 0, /*write_out=*/1);
}